// BiChain_4818953306233
// MI455X (gfx1250) — compile-verified
//
#include <hip/hip_runtime.h>

typedef float v2f __attribute__((ext_vector_type(2)));
typedef float v8f __attribute__((ext_vector_type(8)));

#define Bsz   32768
#define Ddim  1024
#define Cn    40
#define WROW  (Ddim + Cn)       // 1064 floats per weight row
#define NTOT  (2 * Cn)          // 80 combined output columns
#define NT    5                 // five 16-wide N tiles

// ---------------------------------------------------------------------------
// Kernel 1: base[B][80] = src @ [W[:, :1024] ; W_rev[:, :1024]]^T   (fp32 WMMA)
// Block = 256 threads = 8 waves. Each wave register-blocks TWO 16-row M tiles
// (32 rows) x all five 16-col N tiles -> each B fetch feeds 2 WMMAs.
// Per 4-wide K step: 2 A loads + 5 B loads + 10 v_wmma_f32_16x16x4_f32.
// ---------------------------------------------------------------------------
__global__ void __launch_bounds__(256)
bichain_gemm_base(const float* __restrict__ src,
                  const float* __restrict__ W,
                  const float* __restrict__ Wr,
                  float* __restrict__ base)
{
    const int lane = threadIdx.x & 31;
    const int wave = threadIdx.x >> 5;
    const int lo   = lane & 15;          // M row (A) / N col (B) within tile
    const int hi   = lane >> 4;          // selects K pair {0,1} vs {2,3}

    const int mBase = blockIdx.x * 256 + wave * 32;   // 32 rows per wave

    // A fetch pointers: lane reads src[mBase(+16)+lo][k + 2*hi .. +1]
    const float* arow0 = src + (size_t)(mBase + lo) * Ddim + 2 * hi;
    const float* arow1 = arow0 + (size_t)16 * Ddim;

    // B fetch pointers: N-tile nt, lane reads Wcat[nt*16+lo][k + 2*hi .. +1]
    const float* brow[NT];
#pragma unroll
    for (int nt = 0; nt < NT; ++nt) {
        int n = nt * 16 + lo;
        brow[nt] = ((n < Cn) ? (W + (size_t)n * WROW)
                             : (Wr + (size_t)(n - Cn) * WROW)) + 2 * hi;
    }

    v8f acc0[NT] = {};                   // M tile 0 accumulators
    v8f acc1[NT] = {};                   // M tile 1 accumulators

    for (int k0 = 0; k0 < Ddim; k0 += 32) {
        // prefetch src streams ~128 floats ahead (gfx1250 global_prefetch_b8)
        __builtin_prefetch(arow0 + k0 + 128, 0, 1);
        __builtin_prefetch(arow1 + k0 + 128, 0, 1);
#pragma unroll
        for (int kk = 0; kk < 32; kk += 4) {
            const int k = k0 + kk;
            v2f a0 = *(const v2f*)(arow0 + k);
            v2f a1 = *(const v2f*)(arow1 + k);
#pragma unroll
            for (int nt = 0; nt < NT; ++nt) {
                v2f b = *(const v2f*)(brow[nt] + k);
                // 8 args: (neg_a, A, neg_b, B, c_mod, C, reuse_a, reuse_b)
                acc0[nt] = __builtin_amdgcn_wmma_f32_16x16x4_f32(
                    false, a0, false, b, (short)0, acc0[nt], false, false);
                acc1[nt] = __builtin_amdgcn_wmma_f32_16x16x4_f32(
                    false, a1, false, b, (short)0, acc1[nt], false, false);
            }
        }
    }

    // C/D layout: VGPR r, lanes 0-15 -> row r, col lo ; lanes 16-31 -> row r+8
#pragma unroll
    for (int nt = 0; nt < NT; ++nt) {
#pragma unroll
        for (int r = 0; r < 8; ++r) {
            int row0 = mBase + r + 8 * hi;
            base[(size_t)row0 * NTOT + nt * 16 + lo]        = acc0[nt][r];
            base[(size_t)(row0 + 16) * NTOT + nt * 16 + lo] = acc1[nt][r];
        }
    }
}

// ---------------------------------------------------------------------------
// Kernel 2: per-row sequential chain + sigmoid + bidirectional combine.
// One thread per batch row; 40x40 cross-weight blocks staged in LDS.
// ---------------------------------------------------------------------------
__global__ void __launch_bounds__(256)
bichain_chain(const float* __restrict__ base,
              const float* __restrict__ W,
              const float* __restrict__ b,
              const float* __restrict__ Wr,
              const float* __restrict__ br,
              float* __restrict__ out)
{
    __shared__ float cf[Cn * Cn];
    __shared__ float cr[Cn * Cn];
    __shared__ float bfs[Cn];
    __shared__ float brs[Cn];

    for (int idx = threadIdx.x; idx < Cn * Cn; idx += 256) {
        int i = idx / Cn, j = idx - i * Cn;
        cf[idx] = W [(size_t)i * WROW + Ddim + j];
        cr[idx] = Wr[(size_t)i * WROW + Ddim + j];
    }
    if (threadIdx.x < Cn) {
        bfs[threadIdx.x] = b [threadIdx.x];
        brs[threadIdx.x] = br[threadIdx.x];
    }
    __syncthreads();

    const int row = blockIdx.x * 256 + threadIdx.x;
    const float* bp = base + (size_t)row * NTOT;

    float sf[Cn], sr[Cn];
#pragma unroll
    for (int i = 0; i < Cn; ++i) {
        sf[i] = bp[i]      + bfs[i];
        sr[i] = bp[Cn + i] + brs[i];
    }

#pragma unroll
    for (int i = 0; i < Cn; ++i) {
        float zf = sf[i], zr = sr[i];
#pragma unroll
        for (int j = 0; j < i; ++j) {
            zf = __builtin_fmaf(sf[j], cf[i * Cn + j], zf);
            zr = __builtin_fmaf(sr[j], cr[i * Cn + j], zr);
        }
        sf[i] = 1.0f / (1.0f + __expf(-zf));   // overwrite base with score
        sr[i] = 1.0f / (1.0f + __expf(-zr));
    }

    float* op = out + (size_t)row * Cn;
#pragma unroll
    for (int i = 0; i < Cn; ++i)
        op[i] = 0.5f * (sf[i] + sr[Cn - 1 - i]);
}

// ---------------------------------------------------------------------------
extern "C" void kernel_launch(void* const* d_in, const int* in_sizes, int n_in,
                              void* d_out, int out_size, void* d_ws, size_t ws_size,
                              hipStream_t stream)
{
    const float* src  = (const float*)d_in[0];
    // d_in[1] = attn_mask (unused by BiChain)
    const float* W    = (const float*)d_in[2];
    const float* b    = (const float*)d_in[3];
    const float* Wr   = (const float*)d_in[4];
    const float* br   = (const float*)d_in[5];
    float*       out  = (float*)d_out;
    float*       base = (float*)d_ws;          // 32768*80*4 = 10.5 MB scratch

    bichain_gemm_base<<<Bsz / 256, 256, 0, stream>>>(src, W, Wr, base);
    bichain_chain    <<<Bsz / 256, 256, 0, stream>>>(base, W, b, Wr, br, out);
}